// MERAnet_clean_12979391169071
// MI455X (gfx1250) — compile-verified
//
#include <hip/hip_runtime.h>
#include <hip/hip_bf16.h>

typedef __attribute__((ext_vector_type(2))) float v2f;
typedef __attribute__((ext_vector_type(8))) float v8f;

// ---------------------------------------------------------------------------
// Disentangler MPS: bond D=4, 4 sites, out=4. Too small for a 16x16 WMMA tile
// -> scalar VALU, one lane per (batch, patch). In-place on the image interior
// (each patch owns a disjoint 2x2 block).
// ---------------------------------------------------------------------------
__global__ void mera_dis_kernel(float* __restrict__ xbuf,
                                const float* __restrict__ A,   // (P,4,2,4,4)
                                const float* __restrict__ W,   // (P,4,4)
                                const float* __restrict__ h0,  // (P,4)
                                int H, int pd) {
  int idx = blockIdx.x * blockDim.x + threadIdx.x;
  int total = 32 * pd * pd;
  if (idx >= total) return;
  int b = idx & 31;          // lane = batch -> A/W loads wave-uniform (per patch)
  int p = idx >> 5;
  int pi_ = p / pd, pj = p % pd;
  float* img = xbuf + (size_t)b * H * H;
  int r0 = 1 + 2 * pi_, c0 = 1 + 2 * pj;

  float xv[4];
  xv[0] = img[r0 * H + c0];
  xv[1] = img[r0 * H + c0 + 1];
  xv[2] = img[(r0 + 1) * H + c0];
  xv[3] = img[(r0 + 1) * H + c0 + 1];

  float h[4];
#pragma unroll
  for (int d = 0; d < 4; ++d) h[d] = h0[p * 4 + d];

  const float* Ap = A + (size_t)p * 4 * 2 * 16;
#pragma unroll
  for (int s = 0; s < 4; ++s) {
    float phi0 = xv[s], phi1 = 1.0f - xv[s];
    const float* A0 = Ap + s * 2 * 16;
    const float* A1 = A0 + 16;
    float hn[4];
#pragma unroll
    for (int e = 0; e < 4; ++e) {
      float acc = 0.0f;
#pragma unroll
      for (int d = 0; d < 4; ++d)
        acc += h[d] * (phi0 * A0[d * 4 + e] + phi1 * A1[d * 4 + e]);
      hn[e] = fmaxf(acc, 0.0f);
    }
#pragma unroll
    for (int e = 0; e < 4; ++e) h[e] = hn[e];
  }

  const float* Wp = W + (size_t)p * 16;
#pragma unroll
  for (int o = 0; o < 4; ++o) {
    float acc = 0.0f;
#pragma unroll
    for (int d = 0; d < 4; ++d) acc += h[d] * Wp[d * 4 + o];
    img[(r0 + (o >> 1)) * H + (c0 + (o & 1))] = acc;
  }
}

// ---------------------------------------------------------------------------
// Isometry MPS: bond D=16, 4 sites, out=1. One wave per patch.
// h_new = relu(phi0*(H@A0) + phi1*(H@A1)); each H@A is a 32x16x16 f32 GEMM
// done as 2 batch tiles x (chain of 4x V_WMMA_F32_16X16X4_F32).
// H state lives in a per-wave LDS tile ([32][17] padded) and is converted
// between C-layout (WMMA result) and A-operand layout via LDS.
// ---------------------------------------------------------------------------
__global__ void mera_iso_kernel(const float* __restrict__ xin,   // (32,H,H)
                                float* __restrict__ xout,        // (32,H/2,H/2)
                                const float* __restrict__ A,     // (P,4,2,16,16)
                                const float* __restrict__ W,     // (P,16,1)
                                const float* __restrict__ h0,    // (P,16)
                                int H, int pdim, int P) {
  __shared__ float smem[8 * 672];       // 8 waves x (32*17 H-tile + 4*32 phi)
  const int wid = threadIdx.x >> 5;
  const int lane = threadIdx.x & 31;
  const int lrow = lane & 15;
  const int lhi = lane >> 4;
  float* Hl = smem + wid * 672;         // [32][17]
  float* XV = Hl + 544;                 // [4][32] site-major x values

  int p = blockIdx.x * 8 + wid;
  bool active = (p < P);
  int pi_ = 0, pj = 0;
  if (active) {
    pi_ = p / pdim;
    pj = p % pdim;
    const float* img = xin + (size_t)lane * H * H;   // lane = batch
    int r0 = 2 * pi_, c0 = 2 * pj;
    XV[0 * 32 + lane] = img[r0 * H + c0];
    XV[1 * 32 + lane] = img[r0 * H + c0 + 1];
    XV[2 * 32 + lane] = img[(r0 + 1) * H + c0];
    XV[3 * 32 + lane] = img[(r0 + 1) * H + c0 + 1];
#pragma unroll
    for (int d = 0; d < 16; ++d) Hl[lane * 17 + d] = h0[p * 16 + d];
  }
  __syncthreads();

  for (int s = 0; s < 4; ++s) {
    if (active) {
      const float* A0 = A + (((size_t)p * 4 + s) * 2 + 0) * 256;
      const float* A1 = A0 + 256;
      // B operands: B[k][n], k = 4c + 2*lhi + v, n = lrow
      v2f b0[4], b1[4];
#pragma unroll
      for (int c = 0; c < 4; ++c) {
        int k = 4 * c + 2 * lhi;
        b0[c].x = A0[k * 16 + lrow];
        b0[c].y = A0[(k + 1) * 16 + lrow];
        b1[c].x = A1[k * 16 + lrow];
        b1[c].y = A1[(k + 1) * 16 + lrow];
      }
#pragma unroll
      for (int t = 0; t < 2; ++t) {            // batch tiles 0-15, 16-31
        // A operand: A[m][k], m = lrow (batch row in tile), k = 4c+2*lhi+v
        v2f a[4];
#pragma unroll
        for (int c = 0; c < 4; ++c) {
          int row = t * 16 + lrow;
          int k = 4 * c + 2 * lhi;
          a[c].x = Hl[row * 17 + k];
          a[c].y = Hl[row * 17 + k + 1];
        }
        v8f acc0 = {};
        v8f acc1 = {};
#pragma unroll
        for (int c = 0; c < 4; ++c) {
          acc0 = __builtin_amdgcn_wmma_f32_16x16x4_f32(
              false, a[c], false, b0[c], (short)0, acc0, false, false);
          acc1 = __builtin_amdgcn_wmma_f32_16x16x4_f32(
              false, a[c], false, b1[c], (short)0, acc1, false, false);
        }
        // D layout: acc[r] = D[lhi*8 + r][lrow]; combine with per-batch phi
#pragma unroll
        for (int r = 0; r < 8; ++r) {
          int brow = t * 16 + lhi * 8 + r;
          float v = XV[s * 32 + brow];
          float hn = fmaxf(v * acc0[r] + (1.0f - v) * acc1[r], 0.0f);
          Hl[brow * 17 + lrow] = hn;
        }
      }
    }
    __syncthreads();
  }

  if (active) {
    const float* Wp = W + (size_t)p * 16;   // o = 1
    float acc = 0.0f;
#pragma unroll
    for (int d = 0; d < 16; ++d) acc += Hl[lane * 17 + d] * Wp[d];
    xout[(size_t)lane * pdim * pdim + pi_ * pdim + pj] = acc;
  }
}

// ---------------------------------------------------------------------------
// Final classifier MPS: 49 sites, bond 16, out 10, single patch.
// Same WMMA scheme, one wave total.
// ---------------------------------------------------------------------------
__global__ void mera_fin_kernel(const float* __restrict__ xin,  // (32,49)
                                const float* __restrict__ A,    // (49,2,16,16)
                                const float* __restrict__ W,    // (16,10)
                                const float* __restrict__ h0,   // (16)
                                float* __restrict__ out) {      // (32,10)
  __shared__ float Hl[32 * 17];
  __shared__ float XV[49 * 32];
  const int lane = threadIdx.x;     // 32 threads = 1 wave
  const int lrow = lane & 15;
  const int lhi = lane >> 4;

  for (int s = 0; s < 49; ++s) XV[s * 32 + lane] = xin[lane * 49 + s];
#pragma unroll
  for (int d = 0; d < 16; ++d) Hl[lane * 17 + d] = h0[d];
  __syncthreads();

  for (int s = 0; s < 49; ++s) {
    const float* A0 = A + (size_t)s * 2 * 256;
    const float* A1 = A0 + 256;
    v2f b0[4], b1[4];
#pragma unroll
    for (int c = 0; c < 4; ++c) {
      int k = 4 * c + 2 * lhi;
      b0[c].x = A0[k * 16 + lrow];
      b0[c].y = A0[(k + 1) * 16 + lrow];
      b1[c].x = A1[k * 16 + lrow];
      b1[c].y = A1[(k + 1) * 16 + lrow];
    }
#pragma unroll
    for (int t = 0; t < 2; ++t) {
      v2f a[4];
#pragma unroll
      for (int c = 0; c < 4; ++c) {
        int row = t * 16 + lrow;
        int k = 4 * c + 2 * lhi;
        a[c].x = Hl[row * 17 + k];
        a[c].y = Hl[row * 17 + k + 1];
      }
      v8f acc0 = {};
      v8f acc1 = {};
#pragma unroll
      for (int c = 0; c < 4; ++c) {
        acc0 = __builtin_amdgcn_wmma_f32_16x16x4_f32(
            false, a[c], false, b0[c], (short)0, acc0, false, false);
        acc1 = __builtin_amdgcn_wmma_f32_16x16x4_f32(
            false, a[c], false, b1[c], (short)0, acc1, false, false);
      }
#pragma unroll
      for (int r = 0; r < 8; ++r) {
        int brow = t * 16 + lhi * 8 + r;
        float v = XV[s * 32 + brow];
        float hn = fmaxf(v * acc0[r] + (1.0f - v) * acc1[r], 0.0f);
        Hl[brow * 17 + lrow] = hn;
      }
    }
    __syncthreads();
  }

  float hreg[16];
#pragma unroll
  for (int d = 0; d < 16; ++d) hreg[d] = Hl[lane * 17 + d];
#pragma unroll
  for (int o = 0; o < 10; ++o) {
    float acc = 0.0f;
#pragma unroll
    for (int d = 0; d < 16; ++d) acc += hreg[d] * W[d * 10 + o];
    out[lane * 10 + o] = acc;
  }
}

// ---------------------------------------------------------------------------
// d_in layout (setup_inputs order, lists flattened):
//  0: x  | 1-4: A_dis | 5-8: W_dis | 9-12: h_dis
//  13-16: A_iso | 17-20: W_iso | 21-24: h_iso | 25: A_fin | 26: W_fin | 27: h_fin
// ---------------------------------------------------------------------------
extern "C" void kernel_launch(void* const* d_in, const int* in_sizes, int n_in,
                              void* d_out, int out_size, void* d_ws, size_t ws_size,
                              hipStream_t stream) {
  (void)in_sizes; (void)n_in; (void)out_size; (void)ws_size;
  const int Hdims[4] = {112, 56, 28, 14};
  const size_t IMG = (size_t)32 * 112 * 112;

  float* buf0 = (float*)d_ws;
  float* buf1 = buf0 + IMG;

  hipMemcpyAsync(buf0, d_in[0], IMG * sizeof(float),
                 hipMemcpyDeviceToDevice, stream);

  float* cur = buf0;
  float* nxt = buf1;
  for (int j = 0; j < 4; ++j) {
    int H = Hdims[j];
    int pd = (H - 2) / 2;
    int totDis = 32 * pd * pd;
    mera_dis_kernel<<<(totDis + 255) / 256, 256, 0, stream>>>(
        cur, (const float*)d_in[1 + j], (const float*)d_in[5 + j],
        (const float*)d_in[9 + j], H, pd);

    int pi = H / 2;
    int Pi = pi * pi;
    int blocks = (Pi + 7) / 8;   // 8 waves per block, 1 patch per wave
    mera_iso_kernel<<<blocks, 256, 0, stream>>>(
        cur, nxt, (const float*)d_in[13 + j], (const float*)d_in[17 + j],
        (const float*)d_in[21 + j], H, pi, Pi);

    float* t = cur; cur = nxt; nxt = t;
  }

  mera_fin_kernel<<<1, 32, 0, stream>>>(
      cur, (const float*)d_in[25], (const float*)d_in[26],
      (const float*)d_in[27], (float*)d_out);
}